// LearnableSelfLoopLaplacian_62139586839288
// MI455X (gfx1250) — compile-verified
//
#include <hip/hip_runtime.h>
#include <math.h>

// ---------------------------------------------------------------------------
// LearnableSelfLoopLaplacian, N=8192 fp32.  Memory-bound: ~768MB min traffic.
// Pass1: deg partial row-sums via V_WMMA_F32_16X16X4_F32 (A x ones), K-split
//        for occupancy, forward row order to populate L2.
// Pass1b: deterministic combine + rsqrt.
// Pass2: out = I - A' * s[i] * s[j], reverse row order (L2 reuse of A tail),
//        non-temporal b128 loads/stores, nan_to_num semantics.
// ---------------------------------------------------------------------------

typedef __attribute__((ext_vector_type(2))) float v2f;
typedef __attribute__((ext_vector_type(4))) float f4;
typedef __attribute__((ext_vector_type(8))) float v8f;

#define KSEG 512   // columns per K-segment in pass 1
#define NSEG 16    // segments per row  (n / KSEG for n = 8192)

// ---------------------------------------------------------------------------
// Pass 1: partial row sums.  One wave owns a 16-row group x one 512-col
// segment.  D = A_tile(16x4) * ones(4x16) + C accumulates row sums into every
// column of C.  A-tile layout (ISA 7.12.2, 32-bit A 16x4):
//   lanes 0-15 : M = lane,     VGPR0=K0, VGPR1=K1  -> float2 at [row][k..k+1]
//   lanes 16-31: M = lane-16,  VGPR0=K2, VGPR1=K3  -> float2 at [row][k+2..k+3]
// ---------------------------------------------------------------------------
__global__ void __launch_bounds__(256)
rowsum_partial_wmma(const float* __restrict__ A, float* __restrict__ partials, int n)
{
    const int lane      = threadIdx.x & 31;
    const int waveInBlk = threadIdx.x >> 5;
    const int waveId    = blockIdx.x * 8 + waveInBlk;
    const int seg       = waveId & (NSEG - 1);
    const int group     = waveId >> 4;           // which 16-row group
    const int baseRow   = group * 16;
    const int laneRow   = baseRow + (lane & 15);
    const int colOff    = (lane >> 4) * 2;       // 0 or 2

    const float* p = A + (size_t)laneRow * (size_t)n + (size_t)(seg * KSEG + colOff);

    v8f c0 = {}, c1 = {}, c2 = {}, c3 = {};
    v2f ones;  ones[0] = 1.0f;  ones[1] = 1.0f;

    #pragma unroll 2
    for (int k = 0; k < KSEG; k += 16) {
        v2f a0 = *(const v2f*)(p + k);
        v2f a1 = *(const v2f*)(p + k + 4);
        v2f a2 = *(const v2f*)(p + k + 8);
        v2f a3 = *(const v2f*)(p + k + 12);
        c0 = __builtin_amdgcn_wmma_f32_16x16x4_f32(false, a0, false, ones, (short)0, c0, false, false);
        c1 = __builtin_amdgcn_wmma_f32_16x16x4_f32(false, a1, false, ones, (short)0, c1, false, false);
        c2 = __builtin_amdgcn_wmma_f32_16x16x4_f32(false, a2, false, ones, (short)0, c2, false, false);
        c3 = __builtin_amdgcn_wmma_f32_16x16x4_f32(false, a3, false, ones, (short)0, c3, false, false);
    }
    // D layout is elementwise across accumulators -> plain vector adds combine.
    v8f c = (c0 + c1) + (c2 + c3);

    // C/D layout: VGPR r, lanes 0-15 -> M=r ; lanes 16-31 -> M=8+r.
    // Every column holds the same row sum, so lane 0 carries rows base..base+7
    // and lane 16 carries rows base+8..base+15 in c[0..7].
    if ((lane & 15) == 0) {
        const int r0 = baseRow + (lane >> 4) * 8;
        float* dst = partials + (size_t)seg * (size_t)n + (size_t)r0;
        f4 lo, hi;
        lo[0] = c[0]; lo[1] = c[1]; lo[2] = c[2]; lo[3] = c[3];
        hi[0] = c[4]; hi[1] = c[5]; hi[2] = c[6]; hi[3] = c[7];
        *(f4*)(dst)     = lo;
        *(f4*)(dst + 4) = hi;
    }
}

// ---------------------------------------------------------------------------
// Pass 1b: deg[i] = w[i] + sum_seg partials[seg][i];  sinv[i] = 1/sqrt(deg).
// Fixed summation order -> bitwise deterministic (no float atomics).
// ---------------------------------------------------------------------------
__global__ void __launch_bounds__(256)
combine_rsqrt(const float* __restrict__ partials, const float* __restrict__ w,
              float* __restrict__ sinv, int n)
{
    const int i = blockIdx.x * blockDim.x + threadIdx.x;
    if (i >= n) return;
    float deg = w[i];
    #pragma unroll
    for (int s = 0; s < NSEG; ++s)
        deg += partials[(size_t)s * (size_t)n + i];
    sinv[i] = 1.0f / sqrtf(deg);
}

// ---------------------------------------------------------------------------
// Pass 2: out[i][j] = (i==j) - (A[i][j] + (i==j)*w[i]) * s[i] * s[j], then
// nan_to_num.  Rows visited in REVERSE order so the tail of A (still resident
// in the 192MB L2 from pass 1's forward stream) is re-read from L2, not HBM.
// A-loads and out-stores are non-temporal (dead data, don't pollute caches).
// ---------------------------------------------------------------------------
__device__ __forceinline__ float nan2num(float x)
{
    if (isnan(x)) return 0.0f;
    if (isinf(x)) return (x > 0.0f) ? 3.4028234663852886e38f : -3.4028234663852886e38f;
    return x;
}

__global__ void __launch_bounds__(256)
laplacian_scale(const float* __restrict__ A, const float* __restrict__ w,
                const float* __restrict__ sinv, float* __restrict__ out, int n)
{
    const int row = n - 1 - (int)blockIdx.y;            // reverse traversal
    const int j0  = ((int)blockIdx.x * (int)blockDim.x + (int)threadIdx.x) * 4;
    const size_t idx = (size_t)row * (size_t)n + (size_t)j0;

    f4 a  = __builtin_nontemporal_load((const f4*)(A + idx));   // last use of A
    f4 sj = *(const f4*)(sinv + j0);                            // hot 32KB, cached
    const float si = sinv[row];

    f4 o;
    o[0] = -(a[0] * si) * sj[0];
    o[1] = -(a[1] * si) * sj[1];
    o[2] = -(a[2] * si) * sj[2];
    o[3] = -(a[3] * si) * sj[3];

    if (row >= j0 && row < j0 + 4) {                    // diagonal element here
        const int d = row - j0;
        o[d] = 1.0f - (a[d] + w[row]) * si * sj[d];
    }

    o[0] = nan2num(o[0]);
    o[1] = nan2num(o[1]);
    o[2] = nan2num(o[2]);
    o[3] = nan2num(o[3]);

    __builtin_nontemporal_store(o, (f4*)(out + idx));   // streaming store
}

// ---------------------------------------------------------------------------
extern "C" void kernel_launch(void* const* d_in, const int* in_sizes, int n_in,
                              void* d_out, int out_size, void* d_ws, size_t ws_size,
                              hipStream_t stream)
{
    const float* A = (const float*)d_in[0];
    const float* w = (const float*)d_in[1];
    float* out     = (float*)d_out;
    const int n    = in_sizes[1];                 // 8192 (len of self_loop_weight)

    // Scratch layout (floats): [NSEG * n] partial row sums, then [n] sinv.
    float* partials = (float*)d_ws;
    float* sinv     = partials + (size_t)NSEG * (size_t)n;

    // Pass 1: (n/16) row groups x NSEG segments = 8192 waves, 8 waves/block.
    const int wavesTotal = (n / 16) * NSEG;
    rowsum_partial_wmma<<<wavesTotal / 8, 256, 0, stream>>>(A, partials, n);

    // Pass 1b: one thread per row.
    combine_rsqrt<<<(n + 255) / 256, 256, 0, stream>>>(partials, w, sinv, n);

    // Pass 2: 256 threads x 4 floats = 1024 cols per block; grid.y = rows.
    dim3 g2((unsigned)(n / 1024), (unsigned)n, 1);
    laplacian_scale<<<g2, 256, 0, stream>>>(A, w, sinv, out, n);
}